// TreeSupport_39651138076979
// MI455X (gfx1250) — compile-verified
//
#include <hip/hip_runtime.h>

// GNN pipeline: GIN(128->64->64) + GIN(64->32->32) + SAGE(32->1)
// Optimization: GIN's first Linear commutes with segment_sum, so we project
// node features BEFORE the edge scatter (64/32-dim scatters instead of
// 128/64-dim) -- cuts edge gather+atomic traffic ~43%, which is the dominant
// cost (all node tensors are L2-resident on MI455X's 192MB L2).
// Dense math uses v_wmma_f32_16x16x4_f32 (exact fp32 path).

#define N_NODES 50000
#define E_EDGES 600000

typedef __attribute__((ext_vector_type(2))) float v2f;
typedef __attribute__((ext_vector_type(8))) float v8f;

__device__ __forceinline__ v2f ld2g(const float* p) { return *(const v2f*)p; }

// ---------------- zero fill ----------------
__global__ void zero_f32(float* __restrict__ p, int n) {
  int i = blockIdx.x * blockDim.x + threadIdx.x;
  if (i < n) p[i] = 0.0f;
}

// ---------------- scatter-sum: agg[dst[e]][f] += feat[src[e]][f] ----------------
template <int FEAT>
__global__ void scatter_sum(const float* __restrict__ feat,
                            const int* __restrict__ srcIdx,
                            const int* __restrict__ dstIdx,
                            float* __restrict__ agg, int nE) {
  long long t = (long long)blockIdx.x * blockDim.x + threadIdx.x;
  int e = (int)(t / FEAT);
  int f = (int)(t % FEAT);
  if (e >= nE) return;
  int s = srcIdx[e];
  int d = dstIdx[e];
  atomicAdd(&agg[(long long)d * FEAT + f], feat[(long long)s * FEAT + f]);
}

// ---------------- Y = X @ W^T  (per-wave 16-row tile, WMMA f32) ----------------
template <int K, int NOUT>
__global__ __launch_bounds__(128) void gemm_xwt(const float* __restrict__ X,
                                                const float* __restrict__ W,
                                                float* __restrict__ Y) {
  const int lane = threadIdx.x & 31;
  const int wave = threadIdx.x >> 5;
  const int m    = lane & 15;
  const int kh   = lane >> 4;
  const int rowBase = blockIdx.x * 64 + wave * 16;
  const int row  = rowBase + m;
  const int rowc = row < N_NODES ? row : (N_NODES - 1);

  constexpr int NT = NOUT / 16;
  const v8f vzero = {0.f,0.f,0.f,0.f,0.f,0.f,0.f,0.f};
  v8f acc[NT];
#pragma unroll
  for (int nt = 0; nt < NT; ++nt) acc[nt] = vzero;

  const float* xr = X + (long long)rowc * K;
  for (int k0 = 0; k0 < K; k0 += 4) {
    const int kk = k0 + 2 * kh;
    v2f afrag = ld2g(xr + kk);
#pragma unroll
    for (int nt = 0; nt < NT; ++nt) {
      v2f bfrag = ld2g(W + (nt * 16 + m) * K + kk);  // B[k][n] = W[n][k]
      acc[nt] = __builtin_amdgcn_wmma_f32_16x16x4_f32(
          false, afrag, false, bfrag, (short)0, acc[nt], false, false);
    }
  }
#pragma unroll
  for (int nt = 0; nt < NT; ++nt) {
#pragma unroll
    for (int i = 0; i < 8; ++i) {
      int r = rowBase + i + 8 * kh;
      if (r < N_NODES) Y[(long long)r * NOUT + nt * 16 + m] = acc[nt][i];
    }
  }
}

// ---------------- GIN1 tail (fused): u1 = (1+e)y1+agg1+b1a (elementwise),
// h1 = relu(u1@w1b^T + b1b) staged in LDS, y2 = h1@w2a^T ----------------
__global__ __launch_bounds__(128) void gin1_tail_fused(
    const float* __restrict__ y1, const float* __restrict__ agg1,
    const float* __restrict__ b1a, const float* __restrict__ eps1p,
    const float* __restrict__ w1b, const float* __restrict__ b1b,
    const float* __restrict__ w2a, float* __restrict__ y2) {
  __shared__ __align__(16) float h_lds[4][16][68];

  const int lane = threadIdx.x & 31;
  const int wave = threadIdx.x >> 5;
  const int m    = lane & 15;
  const int kh   = lane >> 4;
  const int rowBase = blockIdx.x * 64 + wave * 16;
  const int row  = rowBase + m;
  const int rowc = row < N_NODES ? row : (N_NODES - 1);
  const float epl = 1.0f + eps1p[0];

  const v8f vzero = {0.f,0.f,0.f,0.f,0.f,0.f,0.f,0.f};
  v8f acc[4] = {vzero, vzero, vzero, vzero};

  const float* yr = y1   + (long long)rowc * 64;
  const float* ar = agg1 + (long long)rowc * 64;

  // GEMM: u1(16x64) @ w1b^T(64x64); A frags built elementwise on the fly
  for (int k0 = 0; k0 < 64; k0 += 4) {
    const int kk = k0 + 2 * kh;
    v2f afrag = epl * ld2g(yr + kk) + ld2g(ar + kk) + ld2g(b1a + kk);
#pragma unroll
    for (int nt = 0; nt < 4; ++nt) {
      v2f bfrag = ld2g(w1b + (nt * 16 + m) * 64 + kk);
      acc[nt] = __builtin_amdgcn_wmma_f32_16x16x4_f32(
          false, afrag, false, bfrag, (short)0, acc[nt], false, false);
    }
  }
  // bias + relu -> stage h1 in LDS (D layout: D[i + 8*kh][m])
#pragma unroll
  for (int nt = 0; nt < 4; ++nt) {
    float bias = b1b[nt * 16 + m];
#pragma unroll
    for (int i = 0; i < 8; ++i)
      h_lds[wave][i + 8 * kh][nt * 16 + m] = fmaxf(acc[nt][i] + bias, 0.0f);
  }
  __syncthreads();

  // GEMM: y2 = h1(16x64) @ w2a^T(64x32)
  v8f acc2[2] = {vzero, vzero};
  for (int k0 = 0; k0 < 64; k0 += 4) {
    const int kk = k0 + 2 * kh;
    v2f afrag = *(const v2f*)&h_lds[wave][m][kk];
#pragma unroll
    for (int nt = 0; nt < 2; ++nt) {
      v2f bfrag = ld2g(w2a + (nt * 16 + m) * 64 + kk);
      acc2[nt] = __builtin_amdgcn_wmma_f32_16x16x4_f32(
          false, afrag, false, bfrag, (short)0, acc2[nt], false, false);
    }
  }
#pragma unroll
  for (int nt = 0; nt < 2; ++nt) {
#pragma unroll
    for (int i = 0; i < 8; ++i) {
      int r = rowBase + i + 8 * kh;
      if (r < N_NODES) y2[(long long)r * 32 + nt * 16 + m] = acc2[nt][i];
    }
  }
}

// ---------------- GIN2 tail: u2 = (1+e)y2+agg2+b2a (elementwise),
// h2 = relu(u2@w2b^T + b2b); no LDS needed ----------------
__global__ __launch_bounds__(128) void gin2_tail(
    const float* __restrict__ y2, const float* __restrict__ agg2,
    const float* __restrict__ b2a, const float* __restrict__ eps2p,
    const float* __restrict__ w2b, const float* __restrict__ b2b,
    float* __restrict__ h2) {
  const int lane = threadIdx.x & 31;
  const int wave = threadIdx.x >> 5;
  const int m    = lane & 15;
  const int kh   = lane >> 4;
  const int rowBase = blockIdx.x * 64 + wave * 16;
  const int row  = rowBase + m;
  const int rowc = row < N_NODES ? row : (N_NODES - 1);
  const float epl = 1.0f + eps2p[0];

  const v8f vzero = {0.f,0.f,0.f,0.f,0.f,0.f,0.f,0.f};
  v8f acc[2] = {vzero, vzero};

  const float* yr = y2   + (long long)rowc * 32;
  const float* ar = agg2 + (long long)rowc * 32;

  for (int k0 = 0; k0 < 32; k0 += 4) {
    const int kk = k0 + 2 * kh;
    v2f afrag = epl * ld2g(yr + kk) + ld2g(ar + kk) + ld2g(b2a + kk);
#pragma unroll
    for (int nt = 0; nt < 2; ++nt) {
      v2f bfrag = ld2g(w2b + (nt * 16 + m) * 32 + kk);
      acc[nt] = __builtin_amdgcn_wmma_f32_16x16x4_f32(
          false, afrag, false, bfrag, (short)0, acc[nt], false, false);
    }
  }
#pragma unroll
  for (int nt = 0; nt < 2; ++nt) {
    float bias = b2b[nt * 16 + m];
#pragma unroll
    for (int i = 0; i < 8; ++i) {
      int r = rowBase + i + 8 * kh;
      if (r < N_NODES)
        h2[(long long)r * 32 + nt * 16 + m] = fmaxf(acc[nt][i] + bias, 0.0f);
    }
  }
}

// ---------------- SAGE scatter: weighted messages + degree count ----------------
__global__ void sage_scatter(const float* __restrict__ h2,
                             const int* __restrict__ srcIdx,
                             const int* __restrict__ dstIdx,
                             const float* __restrict__ ew,
                             float* __restrict__ num, float* __restrict__ cnt, int nE) {
  long long t = (long long)blockIdx.x * blockDim.x + threadIdx.x;
  int e = (int)(t >> 5);
  int f = (int)(t & 31);
  if (e >= nE) return;
  int s = srcIdx[e];
  int d = dstIdx[e];
  float w = ew[e];
  atomicAdd(&num[(long long)d * 32 + f], w * h2[(long long)s * 32 + f]);
  if (f == 0) atomicAdd(&cnt[d], 1.0f);
}

// ---------------- SAGE output: relu(mean@wl^T + bl + h2@wr^T) ----------------
__global__ void sage_out(const float* __restrict__ num, const float* __restrict__ cnt,
                         const float* __restrict__ h2,
                         const float* __restrict__ wl, const float* __restrict__ bl,
                         const float* __restrict__ wr, float* __restrict__ out) {
  int i = blockIdx.x * blockDim.x + threadIdx.x;
  if (i >= N_NODES) return;
  float inv = 1.0f / fmaxf(cnt[i], 1.0f);
  const float* np = num + (long long)i * 32;
  const float* hp = h2  + (long long)i * 32;
  float s = 0.0f, r = 0.0f;
#pragma unroll
  for (int f = 0; f < 32; ++f) {
    s += np[f] * wl[f];
    r += hp[f] * wr[f];
  }
  out[i] = fmaxf(s * inv + bl[0] + r, 0.0f);
}

extern "C" void kernel_launch(void* const* d_in, const int* in_sizes, int n_in,
                              void* d_out, int out_size, void* d_ws, size_t ws_size,
                              hipStream_t stream) {
  const float* x    = (const float*)d_in[0];
  const int*   eidx = (const int*)  d_in[1];
  const float* ea   = (const float*)d_in[2];
  const float* w1a  = (const float*)d_in[3];
  const float* b1a  = (const float*)d_in[4];
  const float* w1b  = (const float*)d_in[5];
  const float* b1b  = (const float*)d_in[6];
  const float* eps1 = (const float*)d_in[7];
  const float* w2a  = (const float*)d_in[8];
  const float* b2a  = (const float*)d_in[9];
  const float* w2b  = (const float*)d_in[10];
  const float* b2b  = (const float*)d_in[11];
  const float* eps2 = (const float*)d_in[12];
  const float* wl   = (const float*)d_in[13];
  const float* bl   = (const float*)d_in[14];
  const float* wr   = (const float*)d_in[15];
  float* out = (float*)d_out;

  const int* src = eidx;             // edge_index[0]
  const int* dst = eidx + E_EDGES;   // edge_index[1]

  // workspace layout (floats); agg1..cnt contiguous for one zero-fill
  float* ws   = (float*)d_ws;
  float* y1   = ws;                               // N*64
  float* y2   = y1   + (size_t)N_NODES * 64;      // N*32
  float* h2   = y2   + (size_t)N_NODES * 32;      // N*32
  float* agg1 = h2   + (size_t)N_NODES * 32;      // N*64
  float* agg2 = agg1 + (size_t)N_NODES * 64;      // N*32
  float* num  = agg2 + (size_t)N_NODES * 32;      // N*32
  float* cnt  = num  + (size_t)N_NODES * 32;      // N

  const int TB = 256;
  const int gemmBlocks = (N_NODES + 63) / 64;

  // zero all accumulators in one launch (agg1|agg2|num|cnt contiguous)
  {
    int nz = N_NODES * (64 + 32 + 32 + 1);
    zero_f32<<<(nz + TB - 1) / TB, TB, 0, stream>>>(agg1, nz);
  }

  // GIN1: project first (y1 = x@w1a^T), then 64-dim scatter, then fused tail
  gemm_xwt<128, 64><<<gemmBlocks, 128, 0, stream>>>(x, w1a, y1);
  {
    long long tot = (long long)E_EDGES * 64;
    scatter_sum<64><<<(int)((tot + TB - 1) / TB), TB, 0, stream>>>(y1, src, dst, agg1, E_EDGES);
  }
  gin1_tail_fused<<<gemmBlocks, 128, 0, stream>>>(y1, agg1, b1a, eps1, w1b, b1b, w2a, y2);

  // GIN2: 32-dim scatter on y2, then tail -> h2
  {
    long long tot = (long long)E_EDGES * 32;
    scatter_sum<32><<<(int)((tot + TB - 1) / TB), TB, 0, stream>>>(y2, src, dst, agg2, E_EDGES);
  }
  gin2_tail<<<gemmBlocks, 128, 0, stream>>>(y2, agg2, b2a, eps2, w2b, b2b, h2);

  // SAGE
  {
    long long tot = (long long)E_EDGES * 32;
    sage_scatter<<<(int)((tot + TB - 1) / TB), TB, 0, stream>>>(h2, src, dst, ea, num, cnt, E_EDGES);
    sage_out<<<(N_NODES + TB - 1) / TB, TB, 0, stream>>>(num, cnt, h2, wl, bl, wr, out);
  }
}